// HeteroDotProductPredictor_31361851196214
// MI455X (gfx1250) — compile-verified
//
#include <hip/hip_runtime.h>

// Per-edge dot product: score[e] = sum_k h[src[e]][k] * h[dst[e]][k]
// Memory-bound (AI ~ 0.25 FLOP/byte); h (51.2 MB) is L2-resident on MI455X.
// Strategy: wave32 handles a 16-edge tile.
//   1) stage 16 src rows + 16 dst rows into LDS, one coalesced b128 load per row
//   2) 32x V_WMMA_F32_16X16X4_F32 accumulate S x D^T; diagonal = the 16 scores
//   3) spill 16x16 accumulator tile to LDS, lanes 0..15 pick the diagonal

typedef float v2f __attribute__((ext_vector_type(2)));
typedef float v8f __attribute__((ext_vector_type(8)));

#define D_FEAT     128
#define TILE       16
#define ROWSTRIDE  132   // 128 + 4 dword pad: stride mod 64 == 4 -> conflict-free WMMA feeds
#define WPB        2     // waves per block
#define BLOCK      (WPB * 32)
#define LDS_F_PER_WAVE (2 * TILE * ROWSTRIDE + TILE * TILE)

__global__ __launch_bounds__(BLOCK)
void edge_dot_wmma(const float* __restrict__ h,
                   const int*   __restrict__ src,
                   const int*   __restrict__ dst,
                   float*       __restrict__ out,
                   int n_edges, int n_nodes)
{
    __shared__ float lds[WPB][LDS_F_PER_WAVE];
    const int wave = threadIdx.x >> 5;
    const int lane = threadIdx.x & 31;

    float* srcT = &lds[wave][0];
    float* dstT = srcT + TILE * ROWSTRIDE;
    float* dmat = dstT + TILE * ROWSTRIDE;

    const long long tile = (long long)blockIdx.x * WPB + wave;
    const int e0 = (int)(tile * TILE);

    // ---- 32 node ids for this tile: lanes 0-15 -> src, lanes 16-31 -> dst ----
    int e  = e0 + (lane & 15);
    int ce = (e < n_edges) ? e : 0;              // clamp tail; keep EXEC full
    int node = (lane < 16) ? __builtin_nontemporal_load(src + ce)
                           : __builtin_nontemporal_load(dst + ce);
    if ((unsigned)node >= (unsigned)n_nodes) node = 0;

    // ---- stage 32 feature rows into LDS; one coalesced global b128 per row ----
    #pragma unroll 4
    for (int v = 0; v < 32; ++v) {
        int nv = __shfl(node, v, 32);
        const float4 g = *((const float4*)(h + (size_t)nv * D_FEAT) + lane);
        float* row = (v < TILE) ? (srcT + v * ROWSTRIDE)
                                : (dstT + (v - TILE) * ROWSTRIDE);
        *(float4*)(row + 4 * lane) = g;          // 16B-aligned: 132 dwords = 528 B
    }
    __builtin_amdgcn_wave_barrier();             // same-wave DS ops are in-order in HW

    // ---- WMMA accumulate: D[m][n] = sum_k srcT[m][k] * dstT[n][k] ----
    // 16x4 f32 A layout: lanes 0-15 hold K=k..k+1 of row M=lane,
    //                    lanes 16-31 hold K=k+2..k+3 of row M=lane-16.
    // B = dstT^T has the symmetric per-lane layout.
    const int   r    = lane & 15;
    const int   koff = (lane >> 4) << 1;
    const float* pa  = srcT + r * ROWSTRIDE + koff;
    const float* pb  = dstT + r * ROWSTRIDE + koff;

    v8f acc0 = {0.f, 0.f, 0.f, 0.f, 0.f, 0.f, 0.f, 0.f};
    v8f acc1 = {0.f, 0.f, 0.f, 0.f, 0.f, 0.f, 0.f, 0.f};
    #pragma unroll
    for (int k = 0; k < D_FEAT; k += 8) {        // two chains for ILP
        v2f a0 = *(const v2f*)(pa + k);
        v2f b0 = *(const v2f*)(pb + k);
        acc0 = __builtin_amdgcn_wmma_f32_16x16x4_f32(false, a0, false, b0,
                                                     (short)0, acc0, false, false);
        v2f a1 = *(const v2f*)(pa + k + 4);
        v2f b1 = *(const v2f*)(pb + k + 4);
        acc1 = __builtin_amdgcn_wmma_f32_16x16x4_f32(false, a1, false, b1,
                                                     (short)0, acc1, false, false);
    }
    v8f acc = acc0 + acc1;

    // ---- extract diagonal via tiny LDS tile ----
    // Lane layout of 16x16 f32 D: lane (0-15) -> N=lane, M=vgpr; lane (16-31) -> N=lane-16, M=vgpr+8
    const int mbase = (lane >> 4) * 8;
    const int ncol  = lane & 15;
    #pragma unroll
    for (int rr = 0; rr < 8; ++rr)
        dmat[(mbase + rr) * TILE + ncol] = acc[rr];
    __builtin_amdgcn_wave_barrier();

    if (lane < 16) {
        int eo = e0 + lane;
        if (eo < n_edges)
            __builtin_nontemporal_store(dmat[lane * TILE + lane], out + eo);
    }
}

extern "C" void kernel_launch(void* const* d_in, const int* in_sizes, int n_in,
                              void* d_out, int out_size, void* d_ws, size_t ws_size,
                              hipStream_t stream) {
    const float* h   = (const float*)d_in[0];
    const int*   src = (const int*)d_in[1];   // harness passes integer inputs as int32
    const int*   dst = (const int*)d_in[2];
    float*       out = (float*)d_out;

    const int n_edges = in_sizes[1];
    const int n_nodes = in_sizes[0] / D_FEAT;

    const int ntiles = (n_edges + TILE - 1) / TILE;
    const int blocks = (ntiles + WPB - 1) / WPB;

    edge_dot_wmma<<<blocks, BLOCK, 0, stream>>>(h, src, dst, out, n_edges, n_nodes);
}